// SubtractionGaussian_88923002896506
// MI455X (gfx1250) — compile-verified
//
#include <hip/hip_runtime.h>

// out[b,h,i,j] = ||q_i||^2 + ||k_j||^2 - 2 q_i . k_j
// Q: [BH, 2048, 64] f32; K: [BH, 64, 2048] f32; Out: [BH, 2048, 2048] f32.

typedef __attribute__((ext_vector_type(2))) float v2f;
typedef __attribute__((ext_vector_type(4))) float v4f;
typedef __attribute__((ext_vector_type(8))) float v8f;
typedef __attribute__((ext_vector_type(4))) int   v4i;

#define LQ 2048
#define LK 2048
#define DD 64
#define TM 128          // output tile rows per block
#define TN 128          // output tile cols per block
#define QS_STR 68       // Q tile LDS row stride (floats): conflict-free b64 A-frag loads
#define KS_STR 136      // K tile LDS row stride (floats): disjoint bank ranges per lane-half
#define OB_STR 132      // output staging LDS row stride (floats)

#define AS3 __attribute__((address_space(3)))

union SharedTiles {
  struct {
    float Qs[TM][QS_STR];   // Q tile, unscaled: Qs[m][c]
    float Ks[DD][KS_STR];   // K tile, natural layout: Ks[c][n]
  } in;
  float outb[TM][OB_STR];   // output staging for packed b128 stores
};

__global__ __launch_bounds__(256)
void sqdist_wmma_f32_kernel(const float* __restrict__ Q,
                            const float* __restrict__ K,
                            float* __restrict__ Out) {
  __shared__ SharedTiles sh;
  __shared__ float q2s[TM];
  __shared__ float k2s[TN];

  const int t  = threadIdx.x;
  const int bh = blockIdx.z;
  const int m0 = blockIdx.y * TM;
  const int n0 = blockIdx.x * TN;

  const float* __restrict__ Qb = Q   + (size_t)bh * LQ * DD;
  const float* __restrict__ Kb = K   + (size_t)bh * DD * LK;
  float*       __restrict__ Ob = Out + (size_t)bh * LQ * LK;

  // ---- async copy Q tile (128 x 64) straight into LDS (ASYNCcnt path) ----
  #pragma unroll
  for (int u = 0; u < 8; ++u) {
    int fidx = u * 256 + t;                 // float4 index, 0..2047
    int row  = fidx >> 4;                   // 16 float4 per 64-wide row
    int c4   = (fidx & 15) << 2;
    __builtin_amdgcn_global_load_async_to_lds_b128(
        (v4i*)(Qb + (size_t)(m0 + row) * DD + c4),
        (AS3 v4i*)&sh.in.Qs[row][c4], 0, 0);
  }
  // ---- async copy K tile (64 x 128) straight into LDS ----
  #pragma unroll
  for (int u = 0; u < 8; ++u) {
    int fidx = u * 256 + t;                 // float4 index, 0..2047
    int c    = fidx >> 5;                   // 32 float4 per 128-wide row
    int n4   = (fidx & 31) << 2;
    __builtin_amdgcn_global_load_async_to_lds_b128(
        (v4i*)(Kb + (size_t)c * LK + n0 + n4),
        (AS3 v4i*)&sh.in.Ks[c][n4], 0, 0);
  }
  __builtin_amdgcn_s_wait_asynccnt(0);
  __syncthreads();

  // ---- row norms ----
  if (t < 128) {
    float s = 0.0f;
    #pragma unroll
    for (int c = 0; c < DD; ++c) { float x = sh.in.Qs[t][c]; s += x * x; }
    q2s[t] = s;
  } else {
    int r = t - 128;
    float s = 0.0f;
    #pragma unroll
    for (int c = 0; c < DD; ++c) { float x = sh.in.Ks[c][r]; s += x * x; }
    k2s[r] = s;
  }
  __syncthreads();

  // ---- wave tiling: 2 (M) x 4 (N) waves; each wave: 64x32 = 4x2 tiles of 16x16
  const int wave  = t >> 5;
  const int lane  = t & 31;
  const int waveM = wave >> 2;              // 0..1
  const int waveN = wave & 3;               // 0..3
  const int half  = lane >> 4;              // lane group 0-15 / 16-31
  const int l16   = lane & 15;

  v8f zero = {0.f, 0.f, 0.f, 0.f, 0.f, 0.f, 0.f, 0.f};
  v8f acc[4][2];
  #pragma unroll
  for (int mt = 0; mt < 4; ++mt)
    #pragma unroll
    for (int nt = 0; nt < 2; ++nt)
      acc[mt][nt] = zero;

  // ---- main K loop: 16 steps of K=4, 8 WMMAs per step (acc = +q.k) ----
  // A 16x4 f32 layout: VGPR0 = K=0 (lanes 0-15) / K=2 (lanes 16-31); VGPR1 = K=1 / K=3.
  #pragma unroll
  for (int kk = 0; kk < DD; kk += 4) {
    const int kb = kk + 2 * half;
    v2f a[4], b[2];
    #pragma unroll
    for (int mt = 0; mt < 4; ++mt) {
      int row = waveM * 64 + mt * 16 + l16;
      a[mt] = *(const v2f*)&sh.in.Qs[row][kb];          // ds_load_b64
    }
    #pragma unroll
    for (int nt = 0; nt < 2; ++nt) {
      int col = waveN * 32 + nt * 16 + l16;
      v2f bb;
      bb.x = sh.in.Ks[kb + 0][col];                     // fuse -> ds_load_2addr_b32
      bb.y = sh.in.Ks[kb + 1][col];
      b[nt] = bb;
    }
    #pragma unroll
    for (int mt = 0; mt < 4; ++mt)
      #pragma unroll
      for (int nt = 0; nt < 2; ++nt)
        acc[mt][nt] = __builtin_amdgcn_wmma_f32_16x16x4_f32(
            /*neg_a=*/false, a[mt],
            /*neg_b=*/false, b[nt],
            /*c_mod=*/(short)0, acc[mt][nt],
            /*reuse_a=*/false, /*reuse_b=*/false);
  }

  __syncthreads();   // all waves done reading sh.in before we overwrite it

  // ---- epilogue pass 1: fold q2/k2 and -2 scale, stage into LDS ----
  // C layout: VGPR r -> M = r (lanes 0-15) or r+8 (lanes 16-31); N = lane%16.
  #pragma unroll
  for (int mt = 0; mt < 4; ++mt) {
    int rbase = waveM * 64 + mt * 16 + 8 * half;
    #pragma unroll
    for (int nt = 0; nt < 2; ++nt) {
      int jloc = waveN * 32 + nt * 16 + l16;
      float k2v = k2s[jloc];
      #pragma unroll
      for (int r = 0; r < 8; ++r) {
        int iloc = rbase + r;
        sh.outb[iloc][jloc] = fmaf(-2.0f, acc[mt][nt][r], q2s[iloc] + k2v);
      }
    }
  }
  __syncthreads();

  // ---- epilogue pass 2: packed, non-temporal b128 streaming stores ----
  #pragma unroll
  for (int u = 0; u < 16; ++u) {
    int fidx = u * 256 + t;                 // float4 index over 128x128 tile
    int row  = fidx >> 5;                   // 32 float4 per 128-wide row
    int c4   = (fidx & 31) << 2;
    v4f v = *(const v4f*)&sh.outb[row][c4];
    __builtin_nontemporal_store(v, (v4f*)(Ob + (size_t)(m0 + row) * LK + n0 + c4));
  }
}

extern "C" void kernel_launch(void* const* d_in, const int* in_sizes, int n_in,
                              void* d_out, int out_size, void* d_ws, size_t ws_size,
                              hipStream_t stream) {
  (void)n_in; (void)out_size; (void)d_ws; (void)ws_size;
  const float* Q = (const float*)d_in[0];   // [B,H,Lq,D]
  const float* K = (const float*)d_in[1];   // [B,H,D,Lk]
  float* Out = (float*)d_out;               // [B,H,Lq,Lk]

  const int bh = in_sizes[0] / (LQ * DD);   // B*H (=32 for the reference shapes)
  dim3 grid(LK / TN, LQ / TM, bh);
  sqdist_wmma_f32_kernel<<<grid, dim3(256), 0, stream>>>(Q, K, Out);
}